// NeighborhoodAttentionBlock3D_52321291600548
// MI455X (gfx1250) — compile-verified
//
#include <hip/hip_runtime.h>
#include <math.h>

typedef __attribute__((ext_vector_type(2))) float v2f;
typedef __attribute__((ext_vector_type(8))) float v8f;

#define TT   8
#define HH   24
#define WW   24
#define NPOS (TT * HH * WW)   // 4608
#define CCH  64
#define NHEAD 4
#define DH   16
#define KW   7                // neighborhood size
#define RPB_D 13              // 2*K-1

// ---------------------------------------------------------------------------
// Kernel 1: GroupNorm statistics. 8 groups, each a contiguous 8*4608 slab.
// ---------------------------------------------------------------------------
__global__ void gn_stats_kernel(const float* __restrict__ x,
                                float* __restrict__ stats /* [8][2] */) {
    __shared__ float ssum[256];
    __shared__ float ssq[256];
    const int g   = blockIdx.x;
    const int tid = threadIdx.x;
    const float* p = x + (size_t)g * 8 * NPOS;
    float s = 0.f, q = 0.f;
    for (int i = tid; i < 8 * NPOS; i += 256) {
        float v = p[i];
        s += v;
        q += v * v;
    }
    ssum[tid] = s;
    ssq[tid]  = q;
    __syncthreads();
    for (int off = 128; off > 0; off >>= 1) {
        if (tid < off) {
            ssum[tid] += ssum[tid + off];
            ssq[tid]  += ssq[tid + off];
        }
        __syncthreads();
    }
    if (tid == 0) {
        const float inv_n = 1.0f / (8.0f * NPOS);
        float mu  = ssum[0] * inv_n;
        float var = ssq[0] * inv_n - mu * mu;
        stats[2 * g + 0] = mu;
        stats[2 * g + 1] = rsqrtf(var + 1e-5f);
    }
}

// ---------------------------------------------------------------------------
// Kernel 2: normalize + affine + transpose to position-major Xn[p][c].
// ---------------------------------------------------------------------------
__global__ void gn_norm_kernel(const float* __restrict__ x,
                               const float* __restrict__ stats,
                               const float* __restrict__ gw,
                               const float* __restrict__ gb,
                               float* __restrict__ xn) {
    int idx = blockIdx.x * 256 + threadIdx.x;   // idx = p*64 + c
    if (idx >= NPOS * CCH) return;
    int c = idx & 63;
    int p = idx >> 6;
    int g = c >> 3;
    float mu = stats[2 * g + 0];
    float rs = stats[2 * g + 1];
    xn[idx] = (x[(size_t)c * NPOS + p] - mu) * rs * gw[c] + gb[c];
}

// ---------------------------------------------------------------------------
// Kernel 3: WMMA fp32 GEMM:  C[M, ncols] = A[M,64] * W[ncols,64]^T + bias
// Block = 128 threads (4 waves). Block computes a 16-row x 64-col tile:
// wave w handles cols [blockIdx.y*64 + w*16, +16). K=64 -> 16 x wmma 16x16x4.
// epi==0: row-major store to out[row*ncols + col]
// epi==1: transposed store + residual: out[col*NPOS + row] = v + skip[...]
// ---------------------------------------------------------------------------
__global__ void wmma_gemm_kernel(const float* __restrict__ A,
                                 const float* __restrict__ Wt,   // [ncols][64]
                                 const float* __restrict__ bias, // [ncols]
                                 float* __restrict__ out,
                                 const float* __restrict__ skip,
                                 int ncols, int epi) {
    __shared__ float As[16 * 68];   // 16x64, rows padded to 68 floats
    __shared__ float Bs[64 * 68];   // 64x64, rows padded to 68 floats

    const int m0  = blockIdx.x * 16;
    const int gc  = blockIdx.y * 64;
    const int tid = threadIdx.x;

    // Stage A tile: 1024 floats, 8 per thread.
#pragma unroll
    for (int i = 0; i < 8; ++i) {
        int e = tid * 8 + i;
        int r = e >> 6, c = e & 63;
        As[r * 68 + c] = A[(size_t)(m0 + r) * 64 + c];
    }
    // Stage W slice: 4096 floats, 32 per thread.
#pragma unroll
    for (int i = 0; i < 32; ++i) {
        int e = tid * 32 + i;
        int r = e >> 6, c = e & 63;
        Bs[r * 68 + c] = Wt[(size_t)(gc + r) * 64 + c];
    }
    __syncthreads();

    const int wv   = tid >> 5;
    const int lane = tid & 31;
    const int half = lane >> 4;     // 0: lanes 0-15, 1: lanes 16-31
    const int ml   = lane & 15;     // row within tile (A), col within tile (B)
    const int nl   = wv * 16 + ml;  // local col in Bs

    v8f acc = {0.f, 0.f, 0.f, 0.f, 0.f, 0.f, 0.f, 0.f};

#pragma unroll
    for (int k0 = 0; k0 < 64; k0 += 4) {
        // A 16x4 f32 layout: lanes 0-15 -> K = k0,k0+1 ; lanes 16-31 -> k0+2,k0+3
        v2f a, b;
        int kk = k0 + 2 * half;
        a.x = As[ml * 68 + kk];
        a.y = As[ml * 68 + kk + 1];
        // B 4x16 f32 layout mirrors A's K split; B[k][n] = W[n][k]
        b.x = Bs[nl * 68 + kk];
        b.y = Bs[nl * 68 + kk + 1];
        acc = __builtin_amdgcn_wmma_f32_16x16x4_f32(
            /*neg_a=*/false, a, /*neg_b=*/false, b,
            /*c_mod=*/(short)0, acc, /*reuse_a=*/false, /*reuse_b=*/false);
    }

    const int col = gc + wv * 16 + ml;
    const float bv = bias[col];
#pragma unroll
    for (int r = 0; r < 8; ++r) {
        int row = m0 + r + half * 8;   // D layout: VGPR r = row r / r+8
        float v = acc[r] + bv;
        if (epi == 0) {
            out[(size_t)row * ncols + col] = v;
        } else {
            size_t o = (size_t)col * NPOS + row;
            out[o] = v + skip[o];
        }
    }
}

// ---------------------------------------------------------------------------
// Kernel 4: 3D neighborhood attention. One thread per (position, head).
// qkv row layout: [p][192] = {q[4][16], k[4][16], v[4][16]} by head.
// Online softmax over the 343 neighbors, rpb bias added to scores.
// ---------------------------------------------------------------------------
__global__ void natten_kernel(const float* __restrict__ qkv,
                              const float* __restrict__ rpb, // [4][13][13][13]
                              float* __restrict__ attn_out)  // [p][64]
{
    int idx = blockIdx.x * 256 + threadIdx.x;
    if (idx >= NPOS * NHEAD) return;
    const int head = idx & 3;
    const int p    = idx >> 2;
    const int w = p % WW;
    const int h = (p / WW) % HH;
    const int t = p / (HH * WW);

    const float* qp = qkv + (size_t)p * 192 + head * DH;
    float q[DH];
#pragma unroll
    for (int d = 0; d < DH; ++d) q[d] = qp[d] * 0.25f;   // DH^-0.5

    // start = clamp(i - 3, 0, L - 7)
    const int st = min(max(t - 3, 0), TT - KW);
    const int sh = min(max(h - 3, 0), HH - KW);
    const int sw = min(max(w - 3, 0), WW - KW);

    float m = -INFINITY;
    float l = 0.f;
    float acc[DH];
#pragma unroll
    for (int d = 0; d < DH; ++d) acc[d] = 0.f;

    for (int ai = 0; ai < KW; ++ai) {
        const int tt = st + ai;
        const int rt = tt - t + (KW - 1);
        for (int bi = 0; bi < KW; ++bi) {
            const int hh2 = sh + bi;
            const int rh  = hh2 - h + (KW - 1);
            const float* biasrow =
                rpb + (((size_t)head * RPB_D + rt) * RPB_D + rh) * RPB_D;
            const int pn_base = (tt * HH + hh2) * WW;
            for (int ci = 0; ci < KW; ++ci) {
                const int ww2 = sw + ci;
                const int rw  = ww2 - w + (KW - 1);
                const int pn  = pn_base + ww2;
                const float* kp = qkv + (size_t)pn * 192 + 64 + head * DH;
                float s = biasrow[rw];
#pragma unroll
                for (int d = 0; d < DH; ++d) s += q[d] * kp[d];

                const float mn   = fmaxf(m, s);
                const float corr = __expf(m - mn);   // first iter: exp(-inf)=0
                const float wt   = __expf(s - mn);
                l = l * corr + wt;
                const float* vp = kp + 64;
#pragma unroll
                for (int d = 0; d < DH; ++d)
                    acc[d] = acc[d] * corr + wt * vp[d];
                m = mn;
            }
        }
    }

    const float inv_l = 1.0f / l;
    float* op = attn_out + (size_t)p * 64 + head * DH;
#pragma unroll
    for (int d = 0; d < DH; ++d) op[d] = acc[d] * inv_l;
}

// ---------------------------------------------------------------------------
// Launch
// ---------------------------------------------------------------------------
extern "C" void kernel_launch(void* const* d_in, const int* in_sizes, int n_in,
                              void* d_out, int out_size, void* d_ws, size_t ws_size,
                              hipStream_t stream) {
    const float* x      = (const float*)d_in[0];
    const float* gn_w   = (const float*)d_in[1];
    const float* gn_b   = (const float*)d_in[2];
    const float* qkv_w  = (const float*)d_in[3];
    const float* qkv_b  = (const float*)d_in[4];
    const float* rpb    = (const float*)d_in[5];
    const float* proj_w = (const float*)d_in[6];
    const float* proj_b = (const float*)d_in[7];
    const float* conv_w = (const float*)d_in[8];
    const float* conv_b = (const float*)d_in[9];
    float* out = (float*)d_out;

    float* ws    = (float*)d_ws;
    float* stats = ws;                        // 16 floats
    float* xn    = ws + 64;                   // [4608][64]
    float* qkv   = xn  + (size_t)NPOS * 64;   // [4608][192]
    float* attn  = qkv + (size_t)NPOS * 192;  // [4608][64]
    float* proj  = attn + (size_t)NPOS * 64;  // [4608][64]

    // 1) GroupNorm stats (8 groups)
    gn_stats_kernel<<<8, 256, 0, stream>>>(x, stats);

    // 2) Normalize + transpose to [p][c]
    gn_norm_kernel<<<(NPOS * CCH) / 256, 256, 0, stream>>>(x, stats, gn_w, gn_b, xn);

    // 3) QKV GEMM: [4608,192] = Xn[4608,64] @ qkv_w^T + qkv_b
    {
        dim3 grid(NPOS / 16, 192 / 64);
        wmma_gemm_kernel<<<grid, 128, 0, stream>>>(xn, qkv_w, qkv_b, qkv,
                                                   nullptr, 192, 0);
    }

    // 4) Neighborhood attention
    natten_kernel<<<(NPOS * NHEAD) / 256, 256, 0, stream>>>(qkv, rpb, attn);

    // 5) proj GEMM: [4608,64] = attn @ proj_w^T + proj_b
    {
        dim3 grid(NPOS / 16, 1);
        wmma_gemm_kernel<<<grid, 128, 0, stream>>>(attn, proj_w, proj_b, proj,
                                                   nullptr, 64, 0);
    }

    // 6) conv (1x1) GEMM + transpose-store + residual:
    //    out[c][p] = (proj @ conv_w^T + conv_b)[p][c] + x[c][p]
    {
        dim3 grid(NPOS / 16, 1);
        wmma_gemm_kernel<<<grid, 128, 0, stream>>>(proj, conv_w, conv_b, out,
                                                   x, 64, 1);
    }
}